// LeftPool_68710886801778
// MI455X (gfx1250) — compile-verified
//
#include <hip/hip_runtime.h>
#include <hip/hip_bf16.h>
#include <math.h>

// LeftPool = reverse cumulative max along the innermost axis (W = 128).
// Shape (16, 256, 128, 128) f32 -> 524288 rows x 128 floats.
// Bandwidth bound: 537 MB traffic (268 MB each way), ~23 us roofline at 23.3 TB/s.
//
// Wave32 design: one wave per row, each lane owns a 4-float chunk
// (global_load_b128 NT). In-chunk max tree -> cross-lane exclusive
// suffix-max scan via raw ds_bpermute_b32 with *precomputed clamped
// indices* (clamp-to-31 is neutral for a max scan once lane 31 holds
// -inf), -> serial in-chunk combine. 8 independent rows per wave for
// ILP and preamble amortization. NT hints both ways: 256 MiB working
// set exceeds the 192 MB L2.

typedef float v4f __attribute__((ext_vector_type(4)));

#define WAVES_PER_BLOCK 8
#define ROWS_PER_WAVE 8
#define ROW_W 128

__device__ __forceinline__ float bperm_f32(int idx_bytes, float v) {
    return __builtin_bit_cast(float,
        __builtin_amdgcn_ds_bpermute(idx_bytes, __builtin_bit_cast(int, v)));
}

__global__ __launch_bounds__(WAVES_PER_BLOCK * 32)
void LeftPool_68710886801778_kernel(const float* __restrict__ x,
                                    float* __restrict__ out,
                                    int nrows) {
    const int lane = threadIdx.x & 31;
    const int wave = threadIdx.x >> 5;
    const int row0 = (blockIdx.x * WAVES_PER_BLOCK + wave) * ROWS_PER_WAVE;
    if (row0 >= nrows) return;

    // Clamped shuffle-down source indices (bytes), computed once per wave.
    // min(lane+off, 31): out-of-range lanes read lane 31, which is neutral
    // for the post-shift scan (lane 31 carries -inf).
    const int i1  = min(lane + 1,  31) << 2;
    const int i2  = min(lane + 2,  31) << 2;
    const int i4  = min(lane + 4,  31) << 2;
    const int i8  = min(lane + 8,  31) << 2;
    const int i16 = min(lane + 16, 31) << 2;
    const bool is_last = (lane == 31);

    const size_t base0 = (size_t)row0 * ROW_W + (size_t)lane * 4;

    // ---- burst all loads (s_clause of global_load_b128, NT) ----
    v4f v[ROWS_PER_WAVE];
#pragma unroll
    for (int k = 0; k < ROWS_PER_WAVE; ++k) {
        v[k] = __builtin_nontemporal_load(
            reinterpret_cast<const v4f*>(x + base0 + (size_t)k * ROW_W));
    }

#pragma unroll
    for (int k = 0; k < ROWS_PER_WAVE; ++k) {
        // chunk max (3-op tree)
        const float m = fmaxf(fmaxf(v[k].x, v[k].y), fmaxf(v[k].z, v[k].w));

        // shift down by 1 (exclusive), inject -inf at lane 31
        float e = bperm_f32(i1, m);
        e = is_last ? -INFINITY : e;

        // inclusive suffix-max scan of the shifted values: 5 x (bpermute+max)
        e = fmaxf(e, bperm_f32(i1,  e));
        e = fmaxf(e, bperm_f32(i2,  e));
        e = fmaxf(e, bperm_f32(i4,  e));
        e = fmaxf(e, bperm_f32(i8,  e));
        e = fmaxf(e, bperm_f32(i16, e));
        // e = max over all elements in lanes > lane

        // serial in-chunk reverse combine
        v4f o;
        o.w = fmaxf(v[k].w, e);
        o.z = fmaxf(v[k].z, o.w);
        o.y = fmaxf(v[k].y, o.z);
        o.x = fmaxf(v[k].x, o.y);

        __builtin_nontemporal_store(o,
            reinterpret_cast<v4f*>(out + base0 + (size_t)k * ROW_W));
    }
}

extern "C" void kernel_launch(void* const* d_in, const int* in_sizes, int n_in,
                              void* d_out, int out_size, void* d_ws, size_t ws_size,
                              hipStream_t stream) {
    (void)n_in; (void)out_size; (void)d_ws; (void)ws_size;
    const float* x = (const float*)d_in[0];
    float* out = (float*)d_out;

    const int total = in_sizes[0];          // 67,108,864
    const int nrows = total / ROW_W;        // 524,288 rows
    const int rows_per_block = WAVES_PER_BLOCK * ROWS_PER_WAVE;  // 64
    const int blocks = (nrows + rows_per_block - 1) / rows_per_block;  // 8,192

    LeftPool_68710886801778_kernel<<<dim3(blocks),
                                     dim3(WAVES_PER_BLOCK * 32),
                                     0, stream>>>(x, out, nrows);
}